// LambdaLayer_81372450390546
// MI455X (gfx1250) — compile-verified
//
#include <hip/hip_runtime.h>

typedef __attribute__((ext_vector_type(2))) float v2f;
typedef __attribute__((ext_vector_type(4))) float v4f;
typedef __attribute__((ext_vector_type(8))) float v8f;

#define EPSF 1e-5f
#define NB   16      // batch
#define CIN  256     // dim
#define MM   1024    // H*W
#define DK   16
#define DV   64
#define QC   64      // heads*dim_k

// ---------------------------------------------------------------------------
// Kernel 1: QKV 1x1 convs as f32 WMMA GEMMs  out[o,m] = sum_c W[o,c] x[b,c,m]
// 9 o-tiles of 16: [0..3]=Q (BN fused), [4]=K raw, [5..8]=V (BN fused)
// One wave per (b, o_tile, m_tile): 64 x V_WMMA_F32_16X16X4_F32 over c.
// ---------------------------------------------------------------------------
__global__ __launch_bounds__(256) void qkv_wmma(
    const float* __restrict__ x,  const float* __restrict__ Wq,
    const float* __restrict__ Wk, const float* __restrict__ Wv,
    const float* __restrict__ gq, const float* __restrict__ bq,
    const float* __restrict__ mq, const float* __restrict__ vq,
    const float* __restrict__ gv, const float* __restrict__ bv,
    const float* __restrict__ mv, const float* __restrict__ vv,
    float* __restrict__ Qbuf,   // [b][n][64]  (o = h*16+k fastest)
    float* __restrict__ Kbuf,   // [b][n][16]  (k fastest)
    float* __restrict__ Vbuf)   // [b][n][64]  (v fastest)
{
    const int wave = threadIdx.x >> 5;
    const int lane = threadIdx.x & 31;
    const int task = blockIdx.x * 8 + wave;       // 9216 tasks
    const int b    = task / 576;
    const int rem  = task % 576;
    const int ot   = rem / 64;                    // 0..8
    const int m0   = (rem % 64) * 16;

    const int nL   = lane & 15;                   // A row (M) / B col (N)
    const int half = lane >> 4;                   // K sub-offset selector

    const float* W; int orow;
    if (ot < 4)       { W = Wq; orow = ot * 16; }
    else if (ot == 4) { W = Wk; orow = 0; }
    else              { W = Wv; orow = (ot - 5) * 16; }

    const float* wrow = W + (size_t)(orow + nL) * CIN + half * 2; // A row M=nL
    const float* xcol = x + (size_t)b * CIN * MM + m0 + nL;       // B col N=nL

    v8f acc = {};
    #pragma unroll 4
    for (int c0 = 0; c0 < CIN; c0 += 4) {
        const int ca = c0 + half * 2;
        const v2f a = *reinterpret_cast<const v2f*>(wrow + c0);   // b64 load
        v2f bb;
        bb.x = xcol[(size_t)ca * MM];
        bb.y = xcol[(size_t)(ca + 1) * MM];
        acc = __builtin_amdgcn_wmma_f32_16x16x4_f32(
                  false, a, false, bb, (short)0, acc, false, false);
    }

    const int n = m0 + nL;                        // this lane's column
    if (ot == 4) {
        float* dst = Kbuf + (size_t)(b * MM + n) * DK;
        #pragma unroll
        for (int r = 0; r < 8; ++r) dst[r + half * 8] = acc[r];
    } else if (ot < 4) {
        float* dst = Qbuf + (size_t)(b * MM + n) * QC;
        #pragma unroll
        for (int r = 0; r < 8; ++r) {
            const int o = ot * 16 + r + half * 8;
            const float sc = gq[o] * rsqrtf(vq[o] + EPSF);
            dst[o] = acc[r] * sc + (bq[o] - mq[o] * sc);
        }
    } else {
        float* dst = Vbuf + (size_t)(b * MM + n) * DV;
        #pragma unroll
        for (int r = 0; r < 8; ++r) {
            const int o = (ot - 5) * 16 + r + half * 8;
            const float sc = gv[o] * rsqrtf(vv[o] + EPSF);
            dst[o] = acc[r] * sc + (bv[o] - mv[o] * sc);
        }
    }
}

// ---------------------------------------------------------------------------
// Kernel 2: per-position softmax of K over its 16 channels (in place)
// ---------------------------------------------------------------------------
__global__ __launch_bounds__(256) void ksoftmax(float* __restrict__ Kbuf)
{
    const int i = blockIdx.x * 256 + threadIdx.x;    // 0..16383 = b*1024+n
    float* p = Kbuf + (size_t)i * DK;
    float v[DK], mx = -3.0e38f;
    #pragma unroll
    for (int k = 0; k < DK; ++k) { v[k] = p[k]; mx = fmaxf(mx, v[k]); }
    float s = 0.f;
    #pragma unroll
    for (int k = 0; k < DK; ++k) { v[k] = __expf(v[k] - mx); s += v[k]; }
    const float inv = 1.f / s;
    #pragma unroll
    for (int k = 0; k < DK; ++k) p[k] = v[k] * inv;
}

// ---------------------------------------------------------------------------
// Kernel 3: out[b,n,h,v] = sum_k Q[b,n,h,k] * lam[k,v]
//           lam[k,v]    = sum_m (sK[b,m,k] + emb[n,m,k]) * V[b,m,v]
// emb gathered on the fly from rel_pos_emb[63,63,16] (L2-resident) via the
// closed form  off(m) = 16*m + 496*(m>>5) + C(n,k).
// One wave per (b, n-pair): 2 A-streams share B-fragments; 8 WMMA / K-step.
// ---------------------------------------------------------------------------
__global__ __launch_bounds__(256) void lambda_main(
    const float* __restrict__ rel,    // [63][63][16]
    const float* __restrict__ Qbuf,   // [b][n][64]
    const float* __restrict__ sK,     // [b][m][16]
    const float* __restrict__ Vbuf,   // [b][m][64]
    float* __restrict__ out)          // [b][n][256]  (h*64+v)
{
    __shared__ float slab_all[8][DK * DV];        // 32 KB: one slab per wave

    const int wave = threadIdx.x >> 5;
    const int lane = threadIdx.x & 31;
    const int kM   = lane & 15;                   // A row (k) / shared K idx
    const int half = lane >> 4;

    const int b   = blockIdx.x >> 6;              // 64 blocks per batch
    const int blk = blockIdx.x & 63;
    const int n1  = blk * 16 + wave * 2;
    const int n2  = n1 + 1;

    // emb offset constants: C(n,k) = ((31 - n>>5)*63 + (31 - n&31))*16 + k
    const int C1 = ((31 - (n1 >> 5)) * 63 + (31 - (n1 & 31))) * DK + kM;
    const int C2 = ((31 - (n2 >> 5)) * 63 + (31 - (n2 & 31))) * DK + kM;

    const float* sKb = sK   + (size_t)b * MM * DK + kM;
    const float* Vb  = Vbuf + (size_t)b * MM * DV + kM;

    v8f acc1[4] = {}, acc2[4] = {};

    #pragma unroll 2
    for (int mb = 0; mb < MM; mb += 4) {
        const int jj = (mb & 31) + half * 2;      // even; r1 never crosses 32
        const int e0 = (mb >> 5) * 1008 + jj * 16;
        const int r0 = mb + half * 2;

        const float s0 = sKb[r0 * DK];
        const float s1 = sKb[r0 * DK + DK];
        v2f a1, a2;
        a1.x = rel[e0 + C1]      + s0;
        a1.y = rel[e0 + C1 + 16] + s1;
        a2.x = rel[e0 + C2]      + s0;
        a2.y = rel[e0 + C2 + 16] + s1;

        const float* v0p = Vb + (size_t)r0 * DV;
        #pragma unroll
        for (int t = 0; t < 4; ++t) {
            v2f bv;
            bv.x = v0p[t * 16];
            bv.y = v0p[t * 16 + DV];
            acc1[t] = __builtin_amdgcn_wmma_f32_16x16x4_f32(
                          false, a1, false, bv, (short)0, acc1[t], false, false);
            acc2[t] = __builtin_amdgcn_wmma_f32_16x16x4_f32(
                          false, a2, false, bv, (short)0, acc2[t], false, false);
        }
    }

    // Epilogue: lam -> LDS, then out[h,v] = sum_k Q[h,k] * lam[k,v]
    float* slab = slab_all[wave];
    const int h  = lane >> 3;                     // 0..3
    const int v0 = (lane & 7) * 8;                // 0..56

    #pragma unroll
    for (int pass = 0; pass < 2; ++pass) {
        const v8f* A = pass ? acc2 : acc1;
        #pragma unroll
        for (int t = 0; t < 4; ++t)
            #pragma unroll
            for (int r = 0; r < 8; ++r)
                slab[(r + half * 8) * DV + t * 16 + kM] = A[t][r];
        __syncthreads();

        const int n = pass ? n2 : n1;
        // Q for this lane's head: 16 contiguous floats -> 4 x b128 loads
        const v4f* Qv = reinterpret_cast<const v4f*>(
                            Qbuf + (size_t)(b * MM + n) * QC + h * 16);
        v4f q4[4];
        #pragma unroll
        for (int g = 0; g < 4; ++g) q4[g] = Qv[g];

        v4f olo = {}, ohi = {};
        #pragma unroll
        for (int k = 0; k < DK; ++k) {
            const float q = q4[k >> 2][k & 3];
            const v4f* srow = reinterpret_cast<const v4f*>(slab + k * DV + v0);
            const v4f slo = srow[0], shi = srow[1];    // ds_load_b128 x2
            #pragma unroll
            for (int j = 0; j < 4; ++j) {
                olo[j] = fmaf(q, slo[j], olo[j]);
                ohi[j] = fmaf(q, shi[j], ohi[j]);
            }
        }
        v4f* dst = reinterpret_cast<v4f*>(
                       out + (size_t)(b * MM + n) * 256 + h * 64 + v0);
        dst[0] = olo;                                  // global_store_b128 x2
        dst[1] = ohi;
        __syncthreads();
    }
}

// ---------------------------------------------------------------------------
extern "C" void kernel_launch(void* const* d_in, const int* in_sizes, int n_in,
                              void* d_out, int out_size, void* d_ws, size_t ws_size,
                              hipStream_t stream)
{
    const float* x   = (const float*)d_in[0];
    const float* Wq  = (const float*)d_in[1];
    const float* Wk  = (const float*)d_in[2];
    const float* Wv  = (const float*)d_in[3];
    const float* gq  = (const float*)d_in[4];
    const float* bq  = (const float*)d_in[5];
    const float* mq  = (const float*)d_in[6];
    const float* vq  = (const float*)d_in[7];
    const float* gv  = (const float*)d_in[8];
    const float* bv  = (const float*)d_in[9];
    const float* mv  = (const float*)d_in[10];
    const float* vv  = (const float*)d_in[11];
    const float* rel = (const float*)d_in[12];
    float* out = (float*)d_out;

    float* Qbuf = (float*)d_ws;                        // 4 MB
    float* Vbuf = Qbuf + (size_t)NB * MM * QC;         // 4 MB
    float* Kbuf = Vbuf + (size_t)NB * MM * DV;         // 1 MB

    qkv_wmma<<<1152, 256, 0, stream>>>(x, Wq, Wk, Wv, gq, bq, mq, vq,
                                       gv, bv, mv, vv, Qbuf, Kbuf, Vbuf);
    ksoftmax<<<64, 256, 0, stream>>>(Kbuf);
    lambda_main<<<1024, 256, 0, stream>>>(rel, Qbuf, Kbuf, Vbuf, out);
}